// CameraMetadataAnalyzer_38809324487126
// MI455X (gfx1250) — compile-verified
//
#include <hip/hip_runtime.h>
#include <math.h>

// ---------------------------------------------------------------------------
// CameraMetadataAnalyzer for MI455X (gfx1250, wave32)
//   Kernel 1: per-frame streaming stats (bandwidth-bound; LDS strip-mining,
//             per-wave ds_add_f32 histograms, deterministic reductions)
//   Kernel 2: 3-layer MLP on [8,8] meta via V_WMMA_F32_16X16X4_F32 (exact f32)
// ---------------------------------------------------------------------------

typedef float v2f __attribute__((ext_vector_type(2)));
typedef float v8f __attribute__((ext_vector_type(8)));

#define BATCH 8
#define TT    16
#define CH    3
#define HH    256
#define WW    256
#define STRIP 8
#define HALO  2
#define LROWS (STRIP + 2 * HALO)      // 12
#define NSTRIP (HH / STRIP)           // 32
#define NBINS 256
#define NPIX  (HH * WW)               // 65536
#define NVAL  (CH * NPIX)             // 196608
#define EPSF  1e-6f
#define NTHR  512                     // 16 waves of 32
#define NWAVE (NTHR / 32)

__device__ __forceinline__ int reflect_i(int i, int n) {
    // numpy 'reflect' == cv2 BORDER_REFLECT_101: -1 -> 1, n -> n-2
    if (i < 0) i = -i;
    if (i >= n) i = 2 * n - 2 - i;
    return i;
}

__device__ __forceinline__ float wave_reduce_add(float v) {
#pragma unroll
    for (int off = 16; off > 0; off >>= 1)
        v += __shfl_down(v, off, 32);
    return v;
}

// ---------------------------------------------------------------------------
// Kernel 1: one block per frame, strip-mined through LDS.
//   stats[frame][8] = {brightness, contrast, color_temp, laplacian_var,
//                      saturation, entropy, exposure_var, noise}
// ---------------------------------------------------------------------------
__global__ __launch_bounds__(NTHR) void frame_stats_kernel(
    const float* __restrict__ frames, float* __restrict__ stats)
{
    __shared__ float tile[CH][LROWS][WW];     // 36 KB strip (with reflect halo)
    __shared__ float whist[NWAVE][NBINS];     // 16 KB per-wave histograms
    __shared__ float wred[NWAVE][9];          // per-wave reduction partials
    __shared__ float entterm[NBINS];          // per-bin entropy terms

    const int t     = threadIdx.x;
    const int wave  = t >> 5;
    const int frame = blockIdx.x;
    const float* fb = frames + (size_t)frame * CH * NPIX;

    for (int i = t; i < NWAVE * NBINS; i += NTHR) ((float*)whist)[i] = 0.0f;

    float sum = 0.f, sumsq = 0.f, c0 = 0.f, c1 = 0.f, c2 = 0.f;
    float lsum = 0.f, lsumsq = 0.f, nsum = 0.f, nsumsq = 0.f;
    const float gw0 = 1.f / 16.f, gw1 = 4.f / 16.f, gw2 = 6.f / 16.f;

    for (int s = 0; s < NSTRIP; ++s) {
        const int r0 = s * STRIP;
        __syncthreads();                      // protect LDS reuse (also covers init)

        // Stage strip rows [r0-2, r0+STRIP+2) of all 3 channels into LDS
        for (int idx = t; idx < CH * LROWS * WW; idx += NTHR) {
            int x  = idx & (WW - 1);
            int j  = (idx >> 8) % LROWS;
            int c  = idx / (LROWS * WW);
            int gr = reflect_i(r0 - HALO + j, HH);
            tile[c][j][x] = fb[c * NPIX + gr * WW + x];
        }
        // Prefetch next strip into cache (global_prefetch_b8)
        if (s + 1 < NSTRIP && t < CH * LROWS * 4) {
            int c    = t / (LROWS * 4);
            int rj   = (t % (LROWS * 4)) >> 2;
            int part = t & 3;
            int gr   = reflect_i(r0 + STRIP - HALO + rj, HH);
            __builtin_prefetch(fb + c * NPIX + gr * WW + part * 64, 0, 3);
        }
        __syncthreads();

        for (int p = t; p < STRIP * WW; p += NTHR) {
            const int x  = p & (WW - 1);
            const int ly = (p >> 8) + HALO;
            const float r = tile[0][ly][x];
            const float g = tile[1][ly][x];
            const float b = tile[2][ly][x];

            sum   += r + g + b;
            sumsq += r * r + g * g + b * b;
            c0 += r; c1 += g; c2 += b;

            // 256-bin histogram over [0,1]; integer-valued float adds are exact
            int ir = min(max((int)(r * 256.f), 0), NBINS - 1);
            int ig = min(max((int)(g * 256.f), 0), NBINS - 1);
            int ib = min(max((int)(b * 256.f), 0), NBINS - 1);
            atomicAdd(&whist[wave][ir], 1.0f);   // ds_add_f32
            atomicAdd(&whist[wave][ig], 1.0f);
            atomicAdd(&whist[wave][ib], 1.0f);

            // Laplacian of grayscale (reflect-101 columns, halo rows in LDS)
            const int xm1 = reflect_i(x - 1, WW), xp1 = reflect_i(x + 1, WW);
            const int xm2 = reflect_i(x - 2, WW), xp2 = reflect_i(x + 2, WW);
#define GRAY(j, xx) ((tile[0][j][xx] + tile[1][j][xx] + tile[2][j][xx]) * (1.f / 3.f))
            float lap = GRAY(ly - 1, x) + GRAY(ly + 1, x) +
                        GRAY(ly, xm1) + GRAY(ly, xp1) - 4.f * GRAY(ly, x);
#undef GRAY
            lsum += lap; lsumsq += lap * lap;

            // 5x5 Gaussian (separable [1,4,6,4,1]/16), residual stats
#pragma unroll
            for (int c = 0; c < CH; ++c) {
                float acc = 0.f;
#pragma unroll
                for (int dy = 0; dy < 5; ++dy) {
                    const float* row = tile[c][ly - 2 + dy];
                    float hs = gw0 * row[xm2] + gw1 * row[xm1] + gw2 * row[x] +
                               gw1 * row[xp1] + gw0 * row[xp2];
                    float wy = (dy == 0 || dy == 4) ? gw0 : (dy == 2 ? gw2 : gw1);
                    acc += wy * hs;
                }
                float resv = tile[c][ly][x] - acc;
                nsum += resv; nsumsq += resv * resv;
            }
        }
    }

    // Deterministic reduction: wave shuffles -> per-wave partials -> thread 0
    {
        float vals[9] = {sum, sumsq, c0, c1, c2, lsum, lsumsq, nsum, nsumsq};
#pragma unroll
        for (int k = 0; k < 9; ++k) {
            float rv = wave_reduce_add(vals[k]);
            if ((t & 31) == 0) wred[wave][k] = rv;
        }
    }
    __syncthreads();

    // Histogram merge + per-bin entropy terms
    if (t < NBINS) {
        float cnt = 0.f;
#pragma unroll
        for (int w = 0; w < NWAVE; ++w) cnt += whist[w][t];
        float ph = cnt * (1.0f / (float)NVAL);
        entterm[t] = -(ph * logf(ph + EPSF));
    }
    __syncthreads();

    if (t == 0) {
        float tot[9];
#pragma unroll
        for (int k = 0; k < 9; ++k) {
            float a = 0.f;
            for (int w = 0; w < NWAVE; ++w) a += wred[w][k];
            tot[k] = a;
        }
        float entropy = 0.f;
        for (int i = 0; i < NBINS; ++i) entropy += entterm[i];

        const float invN = 1.0f / (float)NVAL, invP = 1.0f / (float)NPIX;
        float mean     = tot[0] * invN;
        float contrast = sqrtf(fmaxf(tot[1] * invN - mean * mean, 0.f));
        float m0 = tot[2] * invP, m1 = tot[3] * invP, m2 = tot[4] * invP;
        float ctemp = m0 / (m2 + EPSF);
        float mu = (m0 + m1 + m2) * (1.f / 3.f);
        float ev = ((m0 - mu) * (m0 - mu) + (m1 - mu) * (m1 - mu) +
                    (m2 - mu) * (m2 - mu)) * (1.f / 3.f);
        float sat = sqrtf(ev);
        float lm  = tot[5] * invP;
        float lvar = fmaxf(tot[6] * invP - lm * lm, 0.f);
        float nm  = tot[7] * invN;
        float noise = sqrtf(fmaxf(tot[8] * invN - nm * nm, 0.f));

        float* o = stats + frame * 8;
        o[0] = mean; o[1] = contrast; o[2] = ctemp; o[3] = lvar;
        o[4] = sat;  o[5] = entropy;  o[6] = ev;    o[7] = noise;
    }
}

// ---------------------------------------------------------------------------
// Kernel 2: temporal mean + 3-layer MLP via V_WMMA_F32_16X16X4_F32.
// Single wave32. A layout (16x4 f32): lane M=lane&15, VGPRs hold
// K = k0 + 2*half + {0,1}. B layout (4x16): rows striped across lanes.
// C/D layout (16x16 f32, 8 VGPRs): VGPR i <-> M = i + 8*half, N = lane&15.
// ---------------------------------------------------------------------------
__global__ __launch_bounds__(32) void mlp_wmma_kernel(
    const float* __restrict__ stats,
    const float* __restrict__ W1, const float* __restrict__ b1,
    const float* __restrict__ W2, const float* __restrict__ b2,
    const float* __restrict__ W3, const float* __restrict__ b3,
    float* __restrict__ out)
{
    __shared__ float meta[16][8];
    __shared__ float h1s[16][16];
    __shared__ float h2s[16][32];

    const int lane = threadIdx.x;
    const int half = lane >> 4;
    const int lm   = lane & 15;

    for (int i = lane; i < 16 * 8; i += 32) ((float*)meta)[i] = 0.0f;
    __syncthreads();
    for (int e = lane; e < BATCH * 8; e += 32) {   // temporal mean over T
        int bb = e >> 3, k = e & 7;
        float s = 0.f;
        for (int tt = 0; tt < TT; ++tt) s += stats[(bb * TT + tt) * 8 + k];
        meta[bb][k] = s * (1.0f / (float)TT);
    }
    __syncthreads();

    // ---- Layer 1: [16x8] @ [8x16] -> [16x16], relu ----
    v8f acc1 = {};
#pragma unroll
    for (int k0 = 0; k0 < 8; k0 += 4) {
        v2f a, bm;
        a[0]  = meta[lm][k0 + 2 * half + 0];
        a[1]  = meta[lm][k0 + 2 * half + 1];
        bm[0] = W1[(k0 + 2 * half + 0) * 16 + lm];
        bm[1] = W1[(k0 + 2 * half + 1) * 16 + lm];
        acc1 = __builtin_amdgcn_wmma_f32_16x16x4_f32(
            false, a, false, bm, (short)0, acc1, false, false);
    }
    {
        float bias = b1[lm];
#pragma unroll
        for (int i = 0; i < 8; ++i)
            h1s[i + 8 * half][lm] = fmaxf(acc1[i] + bias, 0.0f);
    }
    __syncthreads();

    // ---- Layer 2: [16x16] @ [16x32] -> [16x32], relu ----
#pragma unroll
    for (int nt = 0; nt < 2; ++nt) {
        v8f acc2 = {};
#pragma unroll
        for (int k0 = 0; k0 < 16; k0 += 4) {
            v2f a, bm;
            a[0]  = h1s[lm][k0 + 2 * half + 0];
            a[1]  = h1s[lm][k0 + 2 * half + 1];
            bm[0] = W2[(k0 + 2 * half + 0) * 32 + nt * 16 + lm];
            bm[1] = W2[(k0 + 2 * half + 1) * 32 + nt * 16 + lm];
            acc2 = __builtin_amdgcn_wmma_f32_16x16x4_f32(
                false, a, false, bm, (short)0, acc2, false, false);
        }
        float bias = b2[nt * 16 + lm];
#pragma unroll
        for (int i = 0; i < 8; ++i)
            h2s[i + 8 * half][nt * 16 + lm] = fmaxf(acc2[i] + bias, 0.0f);
    }
    __syncthreads();

    // ---- Layer 3: [16x32] @ [32x32] -> [16x32]; rows 0..7 are valid output ----
#pragma unroll
    for (int nt = 0; nt < 2; ++nt) {
        v8f acc3 = {};
#pragma unroll
        for (int k0 = 0; k0 < 32; k0 += 4) {
            v2f a, bm;
            a[0]  = h2s[lm][k0 + 2 * half + 0];
            a[1]  = h2s[lm][k0 + 2 * half + 1];
            bm[0] = W3[(k0 + 2 * half + 0) * 32 + nt * 16 + lm];
            bm[1] = W3[(k0 + 2 * half + 1) * 32 + nt * 16 + lm];
            acc3 = __builtin_amdgcn_wmma_f32_16x16x4_f32(
                false, a, false, bm, (short)0, acc3, false, false);
        }
        float bias = b3[nt * 16 + lm];
        if (half == 0) {                       // M = i (0..7) valid rows
#pragma unroll
            for (int i = 0; i < 8; ++i)
                out[i * 32 + nt * 16 + lm] = acc3[i] + bias;
        }
    }
}

// ---------------------------------------------------------------------------
extern "C" void kernel_launch(void* const* d_in, const int* in_sizes, int n_in,
                              void* d_out, int out_size, void* d_ws, size_t ws_size,
                              hipStream_t stream) {
    const float* frames = (const float*)d_in[0];
    const float* W1 = (const float*)d_in[1];
    const float* b1 = (const float*)d_in[2];
    const float* W2 = (const float*)d_in[3];
    const float* b2 = (const float*)d_in[4];
    const float* W3 = (const float*)d_in[5];
    const float* b3 = (const float*)d_in[6];
    float* out   = (float*)d_out;
    float* stats = (float*)d_ws;              // [128][8] floats = 4 KB scratch

    frame_stats_kernel<<<BATCH * TT, NTHR, 0, stream>>>(frames, stats);
    mlp_wmma_kernel<<<1, 32, 0, stream>>>(stats, W1, b1, W2, b2, W3, b3, out);
}